// TransE_63677185130911
// MI455X (gfx1250) — compile-verified
//
#include <hip/hip_runtime.h>
#include <cstdint>

#define NUM_ENT        102400
#define NUM_REL        1000
#define DIM            256
#define BATCH          512
#define QT             32                    // queries per LDS tile
#define NGROUP         (BATCH / QT)          // 16
#define ENT_PER_BLOCK  256
#define THREADS        256
#define STAGE_ITERS    ((QT * (DIM / 4)) / THREADS)   // 8 float4 tasks/thread
#define X_BYTES        ((size_t)BATCH * DIM * sizeof(float))  // 512 KB

// Low 32 bits of a generic shared-memory address are the LDS byte offset
// (CDNA5 flat->LDS mapping truncates to addr[31:0]).
__device__ __forceinline__ uint32_t lds_off_of(const void* p) {
    return (uint32_t)(uint64_t)p;
}

// CDNA5 async global -> LDS copy, 16 bytes per active lane (ASYNCcnt path).
__device__ __forceinline__ void async_copy_b128(uint32_t lds_byte_off, const void* gaddr) {
    asm volatile("global_load_async_to_lds_b128 %0, %1, off"
                 :: "v"(lds_byte_off), "v"((uint64_t)gaddr)
                 : "memory");
}

__device__ __forceinline__ void wait_async0() {
#if __has_builtin(__builtin_amdgcn_s_wait_asynccnt)
    __builtin_amdgcn_s_wait_asynccnt(0);
#else
    asm volatile("s_wait_asynccnt 0" ::: "memory");
#endif
}

// Shared inner compute: this thread's entity row vs QT queries staged in x_s.
__device__ __forceinline__ void l1_tile_compute(const float4* __restrict__ e4,
                                                const float*  __restrict__ x_s,
                                                int g, int n,
                                                float* __restrict__ out)
{
    float acc[QT];
#pragma unroll
    for (int q = 0; q < QT; ++q) acc[q] = 0.0f;

    const float4* xs4 = reinterpret_cast<const float4*>(x_s);
    for (int dc = 0; dc < DIM / 8; ++dc) {        // 32 chunks of 8 floats
        float4 ea = e4[dc * 2 + 0];
        float4 eb = e4[dc * 2 + 1];
#pragma unroll
        for (int q = 0; q < QT; ++q) {
            // all lanes read the same address -> LDS broadcast, conflict-free
            float4 xa = xs4[q * (DIM / 4) + dc * 2 + 0];
            float4 xb = xs4[q * (DIM / 4) + dc * 2 + 1];
            float s = acc[q];
            s += fabsf(ea.x - xa.x); s += fabsf(ea.y - xa.y);
            s += fabsf(ea.z - xa.z); s += fabsf(ea.w - xa.w);
            s += fabsf(eb.x - xb.x); s += fabsf(eb.y - xb.y);
            s += fabsf(eb.z - xb.z); s += fabsf(eb.w - xb.w);
            acc[q] = s;
        }
    }
#pragma unroll
    for (int q = 0; q < QT; ++q) {
        out[(size_t)(g * QT + q) * NUM_ENT + n] = -acc[q];
    }
}

// ---------------- Phase 1: x[b] = E[h[b]] + R[r[b]]  (once, into workspace) ----
__global__ __launch_bounds__(THREADS)
void build_queries_kernel(const float* __restrict__ E, const float* __restrict__ R,
                          const int* __restrict__ h_idx, const int* __restrict__ r_idx,
                          float* __restrict__ X)
{
    int t  = blockIdx.x * THREADS + threadIdx.x;   // float4 id, 0..32767
    int b  = t >> 6;
    int dv = t & 63;
    const float4* e4 = reinterpret_cast<const float4*>(E + (size_t)h_idx[b] * DIM);
    const float4* r4 = reinterpret_cast<const float4*>(R + (size_t)r_idx[b] * DIM);
    float4 a = e4[dv];
    float4 c = r4[dv];
    a.x += c.x; a.y += c.y; a.z += c.z; a.w += c.w;
    reinterpret_cast<float4*>(X)[t] = a;
}

// ---------------- Phase 2: main scoring kernel (async-DMA x tiles) ------------
__global__ __launch_bounds__(THREADS)
void transe_l1_main_kernel(const float* __restrict__ E,
                           const float* __restrict__ X,
                           float* __restrict__ out)
{
    __shared__ float x_s[QT * DIM];               // 32 KB query tile

    const int tid = threadIdx.x;
    const int n   = blockIdx.x * ENT_PER_BLOCK + tid;
    const float4* __restrict__ e4 =
        reinterpret_cast<const float4*>(E + (size_t)n * DIM);
    const uint32_t x_base = lds_off_of(x_s);

    for (int g = 0; g < NGROUP; ++g) {
        __syncthreads();  // previous group's readers done before restaging
        // contiguous 32 KB tile -> fully coalesced async DMA into LDS
#pragma unroll
        for (int i = 0; i < STAGE_ITERS; ++i) {
            int t2 = tid + THREADS * i;           // float4 task 0..2047
            async_copy_b128(x_base + (uint32_t)t2 * 16u,
                            X + (size_t)g * QT * DIM + (size_t)t2 * 4);
        }
        wait_async0();
        __syncthreads();

        l1_tile_compute(e4, x_s, g, n, out);
    }
}

// ---------------- Fallback: fused gather+combine (if workspace too small) -----
__global__ __launch_bounds__(THREADS)
void transe_l1_fused_kernel(const float* __restrict__ E,
                            const float* __restrict__ R,
                            const int*   __restrict__ h_idx,
                            const int*   __restrict__ r_idx,
                            float*       __restrict__ out)
{
    __shared__ float xh_s[QT * DIM];
    __shared__ float xr_s[QT * DIM];

    const int tid = threadIdx.x;
    const int n   = blockIdx.x * ENT_PER_BLOCK + tid;
    const float4* __restrict__ e4 =
        reinterpret_cast<const float4*>(E + (size_t)n * DIM);
    const uint32_t xh_base = lds_off_of(xh_s);
    const uint32_t xr_base = lds_off_of(xr_s);

    for (int g = 0; g < NGROUP; ++g) {
        __syncthreads();
#pragma unroll
        for (int i = 0; i < STAGE_ITERS; ++i) {
            int t2 = tid + THREADS * i;
            int q  = t2 >> 6;
            int dv = t2 & 63;
            int hq = h_idx[g * QT + q];
            int rq = r_idx[g * QT + q];
            uint32_t loff = (uint32_t)(q * DIM + dv * 4) * 4u;
            async_copy_b128(xh_base + loff, E + (size_t)hq * DIM + dv * 4);
            async_copy_b128(xr_base + loff, R + (size_t)rq * DIM + dv * 4);
        }
        wait_async0();
        __syncthreads();
        {
            float4*       xh4 = reinterpret_cast<float4*>(xh_s);
            const float4* xr4 = reinterpret_cast<const float4*>(xr_s);
#pragma unroll
            for (int i = 0; i < STAGE_ITERS; ++i) {
                int idx = tid + THREADS * i;
                float4 a = xh4[idx];
                float4 b = xr4[idx];
                a.x += b.x; a.y += b.y; a.z += b.z; a.w += b.w;
                xh4[idx] = a;
            }
        }
        __syncthreads();

        l1_tile_compute(e4, xh_s, g, n, out);
    }
}

extern "C" void kernel_launch(void* const* d_in, const int* in_sizes, int n_in,
                              void* d_out, int out_size, void* d_ws, size_t ws_size,
                              hipStream_t stream) {
    const float* E = (const float*)d_in[0];   // entity_emb   [102400, 256]
    const float* R = (const float*)d_in[1];   // relation_emb [1000, 256]
    const int*   h = (const int*)d_in[2];     // h_idx [512]
    const int*   r = (const int*)d_in[3];     // r_idx [512]
    float*       o = (float*)d_out;           // [512, 102400]

    dim3 block(THREADS);
    dim3 grid(NUM_ENT / ENT_PER_BLOCK);       // 400 blocks

    if (d_ws != nullptr && ws_size >= X_BYTES) {
        float* X = (float*)d_ws;
        // Phase 1: build all 512 query vectors once (32768 float4 tasks).
        build_queries_kernel<<<dim3((BATCH * DIM / 4) / THREADS), block, 0, stream>>>(
            E, R, h, r, X);
        // Phase 2: score against all entities.
        transe_l1_main_kernel<<<grid, block, 0, stream>>>(E, X, o);
    } else {
        transe_l1_fused_kernel<<<grid, block, 0, stream>>>(E, R, h, r, o);
    }
}